// NGCFConv_34419867910501
// MI455X (gfx1250) — compile-verified
//
#include <hip/hip_runtime.h>

// ---------------------------------------------------------------------------
// NGCF conv for MI455X (gfx1250, wave32).
// fp32 WMMA (V_WMMA_F32_16X16X4_F32) for both GEMMs.
// W1/W2 staged in LDS *pair-interleaved* so each B fragment is a single
// ds_load_b64 into an even-aligned VGPR pair (no repack movs).
// feat/st/out are all L2-resident (25.6MB each vs 192MB L2); scatter via
// global_atomic_add_f32. 32-bit unsigned index math keeps addressing in
// GVS (sgpr base + 32-bit voffset) form.
// ---------------------------------------------------------------------------

#define N_NODES 100000
#define N_EDGES 1600000
#define DIM     64
#define SLOPE   0.01f

typedef __attribute__((ext_vector_type(2))) float v2f;
typedef __attribute__((ext_vector_type(8))) float v8f;

// D = A(16x4) * B(4x16) + C(16x16), all fp32.
// A lane holds A[lane&15][k0+2*(lane>>4)+{0,1}] in 2 VGPRs;
// B lane holds B[2*(lane>>4)+{0,1}][lane&15];
// C/D vgpr r: row = r + 8*(lane>>4), col = lane&15.
__device__ __forceinline__ v8f wmma_f32x4(v2f a, v2f b, v8f c) {
    return __builtin_amdgcn_wmma_f32_16x16x4_f32(
        /*neg_a=*/false, a, /*neg_b=*/false, b,
        /*c_mod=*/(short)0, c, /*reuse_a=*/false, /*reuse_b=*/false);
}

// Fill LDS with W pair-interleaved: Wp[p*128 + n*2 + lo] = W[(2p+lo)*64 + n].
// A B-fragment for K-row pair p, column n is then the 8-byte pair at
// Wp + p*128 + n*2  -> one ds_load_b64.
__device__ __forceinline__ void fill_w_pairs(float* __restrict__ Wp,
                                             const float* __restrict__ W) {
    const int t = threadIdx.x;
    #pragma unroll
    for (int i = 0; i < 8; ++i) {
        int pidx = t + 256 * i;          // pair index 0..2047
        int p = pidx >> 6;               // K-row pair 0..31
        int n = pidx & 63;               // column 0..63
        v2f v;
        v.x = W[(2 * p)     * DIM + n];
        v.y = W[(2 * p + 1) * DIM + n];
        *(v2f*)(Wp + (unsigned)(p * 128 + n * 2)) = v;
    }
}

// ---------------------------------------------------------------- degree ----
__global__ void k_zero_deg(unsigned* __restrict__ deg) {
    int i = blockIdx.x * blockDim.x + threadIdx.x;
    if (i < N_NODES) deg[i] = 0u;
}

__global__ void k_count_deg(const int* __restrict__ src, unsigned* __restrict__ deg) {
    int e = blockIdx.x * blockDim.x + threadIdx.x;
    if (e < N_EDGES) atomicAdd(&deg[(unsigned)src[e]], 1u);
}

// ------------------------------------------------ self GEMM: st = feat@W1 ---
__global__ __launch_bounds__(256)
void k_self_gemm(const float* __restrict__ feat,
                 const float* __restrict__ W1,
                 float* __restrict__ st,
                 float* __restrict__ out) {
    __shared__ __align__(16) float Wp[DIM * DIM];
    fill_w_pairs(Wp, W1);
    __syncthreads();

    const int lane = threadIdx.x & 31;
    const int l16  = lane & 15;
    const int half = lane >> 4;
    const int kb   = half * 2;

    const int wid = blockIdx.x * (blockDim.x >> 5) + (threadIdx.x >> 5);
    const int nw  = gridDim.x * (blockDim.x >> 5);
    const int nchunks = N_NODES / 16;   // 6250, exact

    for (int c = wid; c < nchunks; c += nw) {
        const float* arow = feat + (unsigned)((c * 16 + l16) * DIM);
        v8f acc[4] = {};
        #pragma unroll 4
        for (int k0 = 0; k0 < DIM; k0 += 4) {
            v2f a = *(const v2f*)(arow + k0 + kb);
            const float* bp = Wp + (unsigned)(((k0 >> 1) + half) * 128 + l16 * 2);
            #pragma unroll
            for (int ct = 0; ct < 4; ++ct) {
                v2f b = *(const v2f*)(bp + ct * 32);
                acc[ct] = wmma_f32x4(a, b, acc[ct]);
            }
        }
        #pragma unroll
        for (int ct = 0; ct < 4; ++ct) {
            #pragma unroll
            for (int r = 0; r < 8; ++r) {
                unsigned idx = (unsigned)((c * 16 + r + 8 * half) * DIM + ct * 16 + l16);
                float v = acc[ct][r];
                st[idx]  = v;   // gathered later at src
                out[idx] = v;   // embed init = self_term
            }
        }
    }
}

// ------------------------------------------------------------ edge kernel ---
// One wave per 16-edge group: A = f[src]*f[dst] (16x64), B = W2 (LDS),
// msg = (A@W2 + st[src]) * invnorm, atomically added into out[dst].
__global__ __launch_bounds__(256)
void k_edge(const float* __restrict__ feat,
            const float* __restrict__ W2,
            const int* __restrict__ src,
            const int* __restrict__ dst,
            const unsigned* __restrict__ deg,
            const float* __restrict__ st,
            float* __restrict__ out) {
    __shared__ __align__(16) float Wp[DIM * DIM];
    fill_w_pairs(Wp, W2);
    __syncthreads();

    const int lane = threadIdx.x & 31;
    const int l16  = lane & 15;
    const int half = lane >> 4;
    const int kb   = half * 2;

    const int wid = blockIdx.x * (blockDim.x >> 5) + (threadIdx.x >> 5);
    const int nw  = gridDim.x * (blockDim.x >> 5);
    const int ngroups = N_EDGES / 16;   // 100000, exact

    for (int g = wid; g < ngroups; g += nw) {
        const int e0 = g * 16;
        // A-frag sources for this lane's matrix row (edge index = l16)
        const unsigned s = (unsigned)src[e0 + l16];
        const unsigned d = (unsigned)dst[e0 + l16];
        const float* fs = feat + s * (unsigned)DIM;
        const float* fd = feat + d * (unsigned)DIM;

        v8f acc[4] = {};
        #pragma unroll 4
        for (int k0 = 0; k0 < DIM; k0 += 4) {
            v2f as = *(const v2f*)(fs + k0 + kb);
            v2f ad = *(const v2f*)(fd + k0 + kb);
            v2f a  = as * ad;
            const float* bp = Wp + (unsigned)(((k0 >> 1) + half) * 128 + l16 * 2);
            #pragma unroll
            for (int ct = 0; ct < 4; ++ct) {
                v2f b = *(const v2f*)(bp + ct * 32);
                acc[ct] = wmma_f32x4(a, b, acc[ct]);
            }
        }

        // Edge metadata for the rows this lane's C fragment covers:
        // C vgpr r -> edge (r + 8*half) of the group.
        unsigned sO[8], dO[8];    // 32-bit element offsets (fit: < 6.4M)
        float    wR[8];
        #pragma unroll
        for (int r = 0; r < 8; ++r) {
            int er = e0 + r + 8 * half;
            unsigned sn = (unsigned)src[er];
            unsigned dn = (unsigned)dst[er];
            sO[r] = sn * (unsigned)DIM;
            dO[r] = dn * (unsigned)DIM;
            unsigned cs = deg[sn]; cs = cs ? cs : 1u;   // clamp(min=1)
            unsigned cd = deg[dn]; cd = cd ? cd : 1u;
            wR[r] = 1.0f / sqrtf((float)cs * (float)cd);
        }

        #pragma unroll
        for (int ct = 0; ct < 4; ++ct) {
            const unsigned col = (unsigned)(ct * 16 + l16);
            #pragma unroll
            for (int r = 0; r < 8; ++r) {
                float v = (acc[ct][r] + st[sO[r] + col]) * wR[r];
                atomicAdd(out + (dO[r] + col), v);
            }
        }
    }
}

// ------------------------------------------------------------- leaky relu ---
__global__ void k_leaky_relu(float4* __restrict__ out) {
    int i = blockIdx.x * blockDim.x + threadIdx.x;
    if (i < (N_NODES * DIM) / 4) {
        float4 v = out[i];
        v.x = v.x > 0.0f ? v.x : SLOPE * v.x;
        v.y = v.y > 0.0f ? v.y : SLOPE * v.y;
        v.z = v.z > 0.0f ? v.z : SLOPE * v.z;
        v.w = v.w > 0.0f ? v.w : SLOPE * v.w;
        out[i] = v;
    }
}

// ---------------------------------------------------------------- launch ----
extern "C" void kernel_launch(void* const* d_in, const int* in_sizes, int n_in,
                              void* d_out, int out_size, void* d_ws, size_t ws_size,
                              hipStream_t stream) {
    (void)in_sizes; (void)n_in; (void)out_size; (void)ws_size;

    const float* feat = (const float*)d_in[0];
    const float* W1   = (const float*)d_in[1];
    const float* W2   = (const float*)d_in[2];
    const int*   src  = (const int*)d_in[3];
    const int*   dst  = (const int*)d_in[4];
    float* out = (float*)d_out;

    // workspace layout: [deg counts: N u32][pad to 256B][st: N*64 f32]
    unsigned* deg = (unsigned*)d_ws;
    size_t st_off = ((size_t)N_NODES * sizeof(unsigned) + 255) & ~(size_t)255;
    float* st = (float*)((char*)d_ws + st_off);

    k_zero_deg  <<<(N_NODES + 255) / 256, 256, 0, stream>>>(deg);
    k_count_deg <<<(N_EDGES + 255) / 256, 256, 0, stream>>>(src, deg);
    k_self_gemm <<<784,  256, 0, stream>>>(feat, W1, st, out);
    k_edge      <<<2048, 256, 0, stream>>>(feat, W2, src, dst, deg, st, out);
    k_leaky_relu<<<(N_NODES * DIM) / 4 / 256, 256, 0, stream>>>((float4*)out);
}